// AdjacentAttention_27865747817083
// MI455X (gfx1250) — compile-verified
//
#include <hip/hip_runtime.h>
#include <hip/hip_bf16.h>
#include <float.h>

typedef float v2f __attribute__((ext_vector_type(2)));
typedef float v8f __attribute__((ext_vector_type(8)));
typedef float f4  __attribute__((ext_vector_type(4)));

#define N_TOK   8192
#define D_MODEL 256
#define INNER   256
#define KV_COLS 512
#define N_ADJ   32
#define HEADS   4
#define DH      64
#define TOK_TILE 16
#define ATT_SCALE 0.125f   // 64^-0.5

// ---------------------------------------------------------------------------
// Kernel 1: KV = x @ Wkv    ([8192,256] @ [256,512] -> [8192,512] in d_ws)
// One WG = 16 tokens. A-tile staged in LDS, WMMA f32 16x16x4, K=256.
// ---------------------------------------------------------------------------
__global__ __launch_bounds__(256) void kv_proj_kernel(
    const float* __restrict__ x, const float* __restrict__ Wkv,
    float* __restrict__ KV) {
  __shared__ __align__(16) float xs[TOK_TILE * D_MODEL];
  const int t    = threadIdx.x;
  const int tok0 = blockIdx.x * TOK_TILE;

  {  // stage x tile (16 KB) with float4 coalesced loads
    const f4* xg = (const f4*)(x + (size_t)tok0 * D_MODEL);
    f4*       xl = (f4*)xs;
    for (int i = t; i < TOK_TILE * D_MODEL / 4; i += 256) xl[i] = xg[i];
  }
  __syncthreads();

  const int wave  = t >> 5;
  const int lane  = t & 31;
  const int row   = lane & 15;          // M (A) / N (B,D) index
  const int khalf = (lane >> 4) << 1;   // lane half selects K {0,1} vs {2,3}

  for (int nt = wave; nt < KV_COLS / 16; nt += 8) {   // 4 column-tiles / wave
    const int col = nt * 16 + row;
    v8f acc = {0.f, 0.f, 0.f, 0.f, 0.f, 0.f, 0.f, 0.f};
#pragma unroll 8
    for (int k = 0; k < D_MODEL; k += 4) {
      const int kk = k + khalf;
      v2f a = *(const v2f*)(&xs[row * D_MODEL + kk]);      // ds_load_b64
      v2f b;
      b.x = Wkv[(size_t)kk * KV_COLS + col];
      b.y = Wkv[(size_t)(kk + 1) * KV_COLS + col];
      acc = __builtin_amdgcn_wmma_f32_16x16x4_f32(false, a, false, b,
                                                  (short)0, acc, false, false);
    }
    const int orow = tok0 + ((lane >> 4) << 3);            // M = r + 8*(lane/16)
#pragma unroll
    for (int r = 0; r < 8; ++r)
      KV[(size_t)(orow + r) * KV_COLS + col] = acc[r];
  }
}

// ---------------------------------------------------------------------------
// Kernel 2: fused  Q = x@Wq (WMMA) -> gathered masked attention (VALU) ->
//           out = attn_out @ Wout + bout (WMMA) -> d_out
// ---------------------------------------------------------------------------
__global__ __launch_bounds__(256) void fused_attn_kernel(
    const float* __restrict__ x, const int* __restrict__ adj,
    const unsigned char* __restrict__ mask, const float* __restrict__ Wq,
    const float* __restrict__ Wout, const float* __restrict__ bout,
    const float* __restrict__ KV, float* __restrict__ out) {
  __shared__ __align__(16) float xs[TOK_TILE * D_MODEL];     // 16 KB
  __shared__ __align__(16) float qs[TOK_TILE * D_MODEL];     // 16 KB
  __shared__ __align__(16) float outs[TOK_TILE * D_MODEL];   // 16 KB
  __shared__ float sc[TOK_TILE][HEADS][N_ADJ];               // 8 KB
  __shared__ int   idxs[TOK_TILE][N_ADJ];                    // 2 KB

  const int t    = threadIdx.x;
  const int tok0 = blockIdx.x * TOK_TILE;
  const int wave = t >> 5;
  const int lane = t & 31;
  const int row  = lane & 15;
  const int khalf = (lane >> 4) << 1;

  {  // stage x tile
    const f4* xg = (const f4*)(x + (size_t)tok0 * D_MODEL);
    f4*       xl = (f4*)xs;
    for (int i = t; i < TOK_TILE * D_MODEL / 4; i += 256) xl[i] = xg[i];
  }
  __syncthreads();

  // ---- Phase B: Q tile = xs @ Wq  -> qs (LDS) --------------------------
  for (int nt = wave; nt < INNER / 16; nt += 8) {    // 2 column-tiles / wave
    const int col = nt * 16 + row;
    v8f acc = {0.f, 0.f, 0.f, 0.f, 0.f, 0.f, 0.f, 0.f};
#pragma unroll 8
    for (int k = 0; k < D_MODEL; k += 4) {
      const int kk = k + khalf;
      v2f a = *(const v2f*)(&xs[row * D_MODEL + kk]);
      v2f b;
      b.x = Wq[(size_t)kk * INNER + col];
      b.y = Wq[(size_t)(kk + 1) * INNER + col];
      acc = __builtin_amdgcn_wmma_f32_16x16x4_f32(false, a, false, b,
                                                  (short)0, acc, false, false);
    }
    const int lrow = (lane >> 4) << 3;
#pragma unroll
    for (int r = 0; r < 8; ++r) qs[(lrow + r) * D_MODEL + col] = acc[r];
  }
  __syncthreads();

  // ---- Phase C: masked dots (gathered K rows from L2) ------------------
  {
    const int token = t >> 4;          // 16 tokens
    const int sub   = t & 15;          // 2 neighbors per thread
    const int gi    = tok0 + token;
    const f4* qrow  = (const f4*)(qs + token * D_MODEL);
#pragma unroll
    for (int jj = 0; jj < 2; ++jj) {
      const int j  = sub + jj * 16;
      const int ij = adj[(size_t)gi * N_ADJ + j];
      const unsigned char mk = mask[(size_t)gi * N_ADJ + j];
      idxs[token][j] = ij;
      const f4* krow = (const f4*)(KV + (size_t)ij * KV_COLS);
#pragma unroll
      for (int h = 0; h < HEADS; ++h) {
        float s = 0.f;
#pragma unroll
        for (int d4 = 0; d4 < DH / 4; ++d4) {
          f4 kv4 = krow[h * (DH / 4) + d4];
          f4 q4  = qrow[h * (DH / 4) + d4];
          s += q4.x * kv4.x + q4.y * kv4.y + q4.z * kv4.z + q4.w * kv4.w;
        }
        s *= ATT_SCALE;
        sc[token][h][j] = mk ? s : -FLT_MAX;
      }
    }
  }
  __syncthreads();

  // ---- Phase D: softmax over the 32 neighbors (64 rows) ----------------
  if (t < TOK_TILE * HEADS) {
    const int token = t >> 2, h = t & 3;
    float m = -FLT_MAX;
#pragma unroll
    for (int j = 0; j < N_ADJ; ++j) m = fmaxf(m, sc[token][h][j]);
    float p[N_ADJ];
    float ssum = 0.f;
#pragma unroll
    for (int j = 0; j < N_ADJ; ++j) {
      p[j] = expf(sc[token][h][j] - m);   // all-masked row -> uniform, as ref
      ssum += p[j];
    }
    const float inv = 1.f / ssum;
#pragma unroll
    for (int j = 0; j < N_ADJ; ++j) sc[token][h][j] = p[j] * inv;
  }
  __syncthreads();

  // ---- Phase E: attn @ V (gathered V rows) -> outs (LDS) ---------------
  {
    const int token = t >> 4;
    const int l16   = t & 15;          // 16 threads cover 256 cols, stride 16
    float acc[16];
#pragma unroll
    for (int ii = 0; ii < 16; ++ii) acc[ii] = 0.f;
    for (int j = 0; j < N_ADJ; ++j) {
      const int ij = idxs[token][j];
      const float* vrow = KV + (size_t)ij * KV_COLS + INNER;
      float a4[4] = {sc[token][0][j], sc[token][1][j],
                     sc[token][2][j], sc[token][3][j]};
#pragma unroll
      for (int ii = 0; ii < 16; ++ii)
        acc[ii] += a4[ii >> 2] * vrow[l16 + ii * 16];   // col>>6 == ii>>2
    }
#pragma unroll
    for (int ii = 0; ii < 16; ++ii)
      outs[token * D_MODEL + l16 + ii * 16] = acc[ii];
  }
  __syncthreads();

  // ---- Phase F: d_out = outs @ Wout + bout (WMMA) ----------------------
  for (int nt = wave; nt < D_MODEL / 16; nt += 8) {
    const int col = nt * 16 + row;
    v8f acc = {0.f, 0.f, 0.f, 0.f, 0.f, 0.f, 0.f, 0.f};
#pragma unroll 8
    for (int k = 0; k < INNER; k += 4) {
      const int kk = k + khalf;
      v2f a = *(const v2f*)(&outs[row * INNER + kk]);
      v2f b;
      b.x = Wout[(size_t)kk * D_MODEL + col];
      b.y = Wout[(size_t)(kk + 1) * D_MODEL + col];
      acc = __builtin_amdgcn_wmma_f32_16x16x4_f32(false, a, false, b,
                                                  (short)0, acc, false, false);
    }
    const float bias = bout[col];
    const int orow = tok0 + ((lane >> 4) << 3);
#pragma unroll
    for (int r = 0; r < 8; ++r)
      out[(size_t)(orow + r) * D_MODEL + col] = acc[r] + bias;
  }
}

// ---------------------------------------------------------------------------
extern "C" void kernel_launch(void* const* d_in, const int* in_sizes, int n_in,
                              void* d_out, int out_size, void* d_ws,
                              size_t ws_size, hipStream_t stream) {
  (void)in_sizes; (void)n_in; (void)out_size; (void)ws_size;
  const float*         x    = (const float*)d_in[0];
  const int*           adj  = (const int*)d_in[1];
  const unsigned char* mask = (const unsigned char*)d_in[2];
  const float*         Wq   = (const float*)d_in[3];
  const float*         Wkv  = (const float*)d_in[4];
  const float*         Wout = (const float*)d_in[5];
  const float*         bout = (const float*)d_in[6];
  float*               out  = (float*)d_out;
  float*               KV   = (float*)d_ws;   // 8192*512 fp32 = 16 MB scratch

  dim3 grid(N_TOK / TOK_TILE);   // 512 workgroups
  kv_proj_kernel<<<grid, 256, 0, stream>>>(x, Wkv, KV);
  fused_attn_kernel<<<grid, 256, 0, stream>>>(x, adj, mask, Wq, Wout, bout, KV,
                                              out);
}